// PhysicsOnlyBaseline_13907104104483
// MI455X (gfx1250) — compile-verified
//
#include <hip/hip_runtime.h>
#include <stdint.h>

// Bloch-equation RK4 integrator, B independent 3-state systems, T time points.
// Output layout: out[t][i][c], t in [0,T), i in [0,B), c in {u,v,w}.
//
// MI455X reasoning:
//  - 402 MB of write-once output  => store-bandwidth bound (~17us @ 23.3 TB/s)
//  - per-system distinct 3x3 generator => WMMA (one shared matrix per wave)
//    cannot help; VALU RK4 (~1.5 Gops total) hides under the store stream.
//  - CDNA5-specific paths used: GLOBAL_LOAD_ASYNC_TO_LDS_B128 + s_wait_asynccnt
//    to stage t_span into LDS, and TH=NT streaming stores for the output.

#define MAX_T 1024   // LDS staging capacity for t_span (reference uses T=512)

__global__ __launch_bounds__(256) void bloch_rk4_kernel(
    const float* __restrict__ y0,
    const float* __restrict__ tspan,
    const float* __restrict__ params,
    float* __restrict__ out,
    int nB, int nT)
{
    __shared__ __align__(16) float s_t[MAX_T];

    const int tid = threadIdx.x;

    // ---- Stage t_span (uniform, tiny) into LDS with CDNA5 async copy ----
    // Each participating lane copies one 16B chunk from byte offset tid*16 of global tspan to the same offset in s_t.
    const int nChunks = nT >> 2;               // 16B chunks (128 for T=512)
    const uint64_t gbase = (uint64_t)(uintptr_t)tspan;
    if (tid < nChunks) {
        uint32_t lds_off = (uint32_t)(uintptr_t)(&s_t[0]) + (uint32_t)tid * 16u;
        uint32_t g_off   = (uint32_t)tid * 16u;
        asm volatile("global_load_async_to_lds_b128 %0, %1, %2"
                     :
                     : "v"(lds_off), "v"(g_off), "s"(gbase)
                     : "memory");
    }
    // Tail (nT not a multiple of 4) via plain path — not taken for T=512.
    if ((nT & 3) && tid < (nT & 3)) {
        s_t[(nT & ~3) + tid] = tspan[(nT & ~3) + tid];
    }
    asm volatile("s_wait_asynccnt 0" ::: "memory");
    __syncthreads();

    const int i = blockIdx.x * blockDim.x + tid;   // system id
    if (i >= nB) return;

    // ---- Load per-system state and parameters (one-time, cached reads) ----
    const float* yp = y0 + (size_t)i * 3;
    float u = yp[0];
    float v = yp[1];
    float w = yp[2];
    const float* pp = params + (size_t)i * 3;
    const float Om = pp[0];          // params[:,1] is unused by the reference RHS
    const float g  = pp[2];
    const float n2g = -2.0f * g;

    const size_t stride = (size_t)nB * 3;          // floats between time rows
    float* op = out + (size_t)i * 3;

    // ---- t = 0 row: y0, streamed non-temporally ----
    __builtin_nontemporal_store(u, op + 0);
    __builtin_nontemporal_store(v, op + 1);
    __builtin_nontemporal_store(w, op + 2);

    // ---- 511 RK4 steps; dt read per step from LDS (fp32-faithful dts) ----
    float tprev = s_t[0];
    for (int s = 1; s < nT; ++s) {
        const float tcur = s_t[s];
        const float dt   = tcur - tprev;
        tprev = tcur;
        const float hd = 0.5f * dt;

        // k1 = f(y)
        const float k1u = -g * u;
        const float k1v = fmaf(-Om, w, -g * v);
        const float k1w = fmaf(Om, v, fmaf(n2g, w, n2g));

        // k2 = f(y + 0.5*dt*k1)
        const float u2 = fmaf(hd, k1u, u);
        const float v2 = fmaf(hd, k1v, v);
        const float w2 = fmaf(hd, k1w, w);
        const float k2u = -g * u2;
        const float k2v = fmaf(-Om, w2, -g * v2);
        const float k2w = fmaf(Om, v2, fmaf(n2g, w2, n2g));

        // k3 = f(y + 0.5*dt*k2)
        const float u3 = fmaf(hd, k2u, u);
        const float v3 = fmaf(hd, k2v, v);
        const float w3 = fmaf(hd, k2w, w);
        const float k3u = -g * u3;
        const float k3v = fmaf(-Om, w3, -g * v3);
        const float k3w = fmaf(Om, v3, fmaf(n2g, w3, n2g));

        // k4 = f(y + dt*k3)
        const float u4 = fmaf(dt, k3u, u);
        const float v4 = fmaf(dt, k3v, v);
        const float w4 = fmaf(dt, k3w, w);
        const float k4u = -g * u4;
        const float k4v = fmaf(-Om, w4, -g * v4);
        const float k4w = fmaf(Om, v4, fmaf(n2g, w4, n2g));

        // y += dt/6 * (k1 + 2*k2 + 2*k3 + k4)
        const float c = dt * (1.0f / 6.0f);
        u = fmaf(c, fmaf(2.0f, k2u, k1u) + fmaf(2.0f, k3u, k4u), u);
        v = fmaf(c, fmaf(2.0f, k2v, k1v) + fmaf(2.0f, k3v, k4v), v);
        w = fmaf(c, fmaf(2.0f, k2w, k1w) + fmaf(2.0f, k3w, k4w), w);

        op += stride;
        __builtin_nontemporal_store(u, op + 0);
        __builtin_nontemporal_store(v, op + 1);
        __builtin_nontemporal_store(w, op + 2);
    }
}

extern "C" void kernel_launch(void* const* d_in, const int* in_sizes, int n_in,
                              void* d_out, int out_size, void* d_ws, size_t ws_size,
                              hipStream_t stream) {
    const float* y0     = (const float*)d_in[0];   // (B,3) f32
    const float* tspan  = (const float*)d_in[1];   // (T,)  f32
    const float* params = (const float*)d_in[2];   // (B,3) f32
    float* out = (float*)d_out;                    // (T,B,3) f32

    int nB = in_sizes[0] / 3;
    int nT = in_sizes[1];
    if (nT > MAX_T) nT = MAX_T;    // LDS staging bound (reference: T=512)

    const int block = 256;
    const int grid  = (nB + block - 1) / block;
    bloch_rk4_kernel<<<grid, block, 0, stream>>>(y0, tspan, params, out, nB, nT);
}